// LSTM_noforget_43430709297557
// MI455X (gfx1250) — compile-verified
//
#include <hip/hip_runtime.h>
#include <stdint.h>

// ---------------------------------------------------------------------------
// LSTM (no-forget) for MI455X / gfx1250, wave32.
//   B=64, S=2048, I=256, H=512, 3H=1536.
// gates_t = [x_t | h_{t-1}] @ [W;U] + bias   (K = 768, fused input projection)
// Persistent kernel: 16 WGs x 256 threads, one grid barrier per timestep.
// Matrix math: V_WMMA_F32_16X16X4_F32 (exact fp32 on the matrix pipe).
// Staging: GLOBAL_LOAD_ASYNC_TO_LDS_B128 (ASYNCcnt) — x is double-buffered one
// step ahead; h DMA overlaps with the x-part of the K loop.
// ---------------------------------------------------------------------------

#define S_LEN   2048
#define BATCH   64
#define IN_DIM  256
#define HID     512
#define G3      1536
#define KSTEPS  192          // (IN_DIM + HID) / 4
#define KS_X    64           // IN_DIM / 4
#define KS_H    128          // HID / 4
#define NWG     16
#define HC      32           // H columns per workgroup (512 / 16)
#define NTILES  6            // (3*HC)/16 N-tiles per WG
#define XSTRIDE 260          // 256 + 4 pad: 260 % 64 == 4 (bank spread), % 4 == 0 (b128)
#define HSTRIDE 516          // 512 + 4 pad: same properties
#define GSTRIDE 100          // 96 + 4 pad for gate-exchange buffer
#define XBUFSZ  (BATCH * XSTRIDE)

typedef __attribute__((ext_vector_type(2))) float v2f;
typedef __attribute__((ext_vector_type(8))) float v8f;

#if __has_builtin(__builtin_amdgcn_global_load_async_to_lds_b128) && \
    __has_builtin(__builtin_amdgcn_s_wait_asynccnt)
#define HAVE_ASYNC_LDS 1
typedef int v4i_ __attribute__((ext_vector_type(4)));
typedef __attribute__((address_space(1))) v4i_ g_v4i;   // global 16B packet
typedef __attribute__((address_space(3))) v4i_ l_v4i;   // LDS 16B packet
// 16-byte global -> LDS async copy (tracked by ASYNCcnt, in-order completion).
__device__ __forceinline__ void async_cp16(const float* gsrc, float* ldst) {
    __builtin_amdgcn_global_load_async_to_lds_b128(
        (g_v4i*)(void*)gsrc, (l_v4i*)(void*)ldst, 0, 0);
}
#else
#define HAVE_ASYNC_LDS 0
#endif

__device__ __forceinline__ float fsigmoid(float x) {
    return 1.0f / (1.0f + __expf(-x));           // v_exp_f32 (trans pipe)
}
__device__ __forceinline__ float ftanh(float x) {
    return 1.0f - 2.0f / (__expf(2.0f * x) + 1.0f);
}

// Global gate column for (workgroup grp, local column colLocal in [0,96)).
__device__ __host__ __forceinline__ int gate_col(int grp, int colLocal) {
    int gate   = colLocal >> 5;          // 0:i  1:g  2:o
    int within = colLocal & 31;
    return gate * HID + grp * HC + within;
}

// ---------------------------------------------------------------------------
// Pre-pack B = [W;U] (768 x 1536) into per-lane WMMA fragment order.
//   B[k0][col], B[k0+1][col],  k0 = 4*kstep + (lane<16 ? 0 : 2)
// Flat layout: Bpack[(((grp*6 + nt)*192 + ks)*32 + lane)*2 + {0,1}]
// ---------------------------------------------------------------------------
__global__ void pack_B_kernel(const float* __restrict__ W,
                              const float* __restrict__ U,
                              float* __restrict__ Bpack) {
    int idx = blockIdx.x * blockDim.x + threadIdx.x;
    const int TOT = NWG * NTILES * KSTEPS * 32;
    if (idx >= TOT) return;
    int lane = idx & 31;
    int ks   = (idx >> 5) % KSTEPS;
    int nt   = (idx / (32 * KSTEPS)) % NTILES;
    int grp  = idx / (32 * KSTEPS * NTILES);

    int colLocal = nt * 16 + (lane & 15);
    int col      = gate_col(grp, colLocal);
    int k0       = ks * 4 + ((lane < 16) ? 0 : 2);

    float v0 = (k0 < IN_DIM) ? W[(size_t)k0 * G3 + col]
                             : U[(size_t)(k0 - IN_DIM) * G3 + col];
    float v1 = (k0 + 1 < IN_DIM) ? W[(size_t)(k0 + 1) * G3 + col]
                                 : U[(size_t)(k0 + 1 - IN_DIM) * G3 + col];
    Bpack[(size_t)idx * 2 + 0] = v0;
    Bpack[(size_t)idx * 2 + 1] = v1;
}

// ---------------------------------------------------------------------------
// Grid-wide barrier: counter-based, generation = step index.
// ---------------------------------------------------------------------------
__device__ __forceinline__ void grid_sync(unsigned* cnt, int step) {
    __syncthreads();
    if (threadIdx.x == 0) {
        __hip_atomic_fetch_add(cnt, 1u, __ATOMIC_RELEASE, __HIP_MEMORY_SCOPE_AGENT);
        unsigned target = (unsigned)(step + 1) * (unsigned)NWG;
        while (__hip_atomic_load(cnt, __ATOMIC_ACQUIRE, __HIP_MEMORY_SCOPE_AGENT) < target) {
            __builtin_amdgcn_s_sleep(1);
        }
    }
    __syncthreads();
}

// ---------------------------------------------------------------------------
// Persistent LSTM kernel. grid = 16 WGs, block = 256 (8 wave32).
// Wave w: M-tile m = w&3 (batch rows 16m..16m+15), N-tiles 3*(w>>2)..+2.
// ---------------------------------------------------------------------------
__global__ void __launch_bounds__(256)
lstm_noforget_kernel(const float* __restrict__ x,
                     const float* __restrict__ bias,
                     const float* __restrict__ Bpack,
                     float* __restrict__ hbuf,      // [2][64][512] double buffer
                     unsigned* __restrict__ cnt,    // grid barrier counter
                     float* __restrict__ out_hs,    // [64][2048][512]
                     float* __restrict__ out_hf,    // [64][512]
                     float* __restrict__ out_cf) {  // [64][512]
    extern __shared__ float smem[];
    float* xA     = smem;                        // [2][64][XSTRIDE]
    float* hA     = smem + 2 * XBUFSZ;           // [64][HSTRIDE]
    float* gatesL = hA + BATCH * HSTRIDE;        // [64][GSTRIDE]

    const int grp  = blockIdx.x;
    const int tid  = threadIdx.x;
    const int lane = tid & 31;
    const int wv   = tid >> 5;
    const int m0   = (wv & 3) * 16;        // batch-row tile base
    const int ntb  = (wv >> 2) * 3;        // first N-tile of this wave
    const int lrow = lane & 15;
    const int koff = (lane < 16) ? 0 : 2;  // ISA A-frag: lanes 16-31 hold K+2

    // staging coords: 4 threads per batch row
    const int sb = tid >> 2;               // 0..63
    const int sq = tid & 3;                // 0..3

    // Per-lane bias for each of this wave's 3 gate-column tiles (loop-invariant).
    const float bias0 = bias[gate_col(grp, (ntb + 0) * 16 + lrow)];
    const float bias1 = bias[gate_col(grp, (ntb + 1) * 16 + lrow)];
    const float bias2 = bias[gate_col(grp, (ntb + 2) * 16 + lrow)];

    // Loop-invariant fragment bases.
    const float* aHBase = hA + (m0 + lrow) * HSTRIDE + koff;
    const size_t bStep  = (size_t)KSTEPS * 64;
    const float* bBase0 = Bpack + (size_t)(grp * NTILES + ntb + 0) * bStep + 2 * lane;
    const float* bBase1 = Bpack + (size_t)(grp * NTILES + ntb + 1) * bStep + 2 * lane;
    const float* bBase2 = Bpack + (size_t)(grp * NTILES + ntb + 2) * bStep + 2 * lane;

#if HAVE_ASYNC_LDS
    // Prologue: async-stage x_0 into LDS buffer 0 (16 x b128 per thread).
    {
        const float* xs = x + ((size_t)sb * S_LEN + 0) * IN_DIM + sq * 64;
        float* xd = xA + sb * XSTRIDE + sq * 64;
        #pragma unroll
        for (int i = 0; i < 16; ++i) async_cp16(xs + 4 * i, xd + 4 * i);
    }
#endif

    int cur = 0;  // h double-buffer index (hbuf[0] zeroed by host-side memset)

    for (int t = 0; t < S_LEN; ++t) {
#if HAVE_ASYNC_LDS
        // Issue h_{t-1} DMA (32 x b128/thread) — overlaps with x-part K loop.
        {
            const float* hs = hbuf + (size_t)cur * (BATCH * HID) + sb * HID + sq * 128;
            float* hd = hA + sb * HSTRIDE + sq * 128;
            #pragma unroll
            for (int i = 0; i < 32; ++i) async_cp16(hs + 4 * i, hd + 4 * i);
        }
        // Issue x_{t+1} DMA one step ahead, then wait only for x_t.
        // Per-thread outstanding (in issue order): x_t(16), h(32), x_{t+1}(16).
        if (t + 1 < S_LEN) {
            const float* xs = x + ((size_t)sb * S_LEN + (t + 1)) * IN_DIM + sq * 64;
            float* xd = xA + (size_t)((t + 1) & 1) * XBUFSZ + sb * XSTRIDE + sq * 64;
            #pragma unroll
            for (int i = 0; i < 16; ++i) async_cp16(xs + 4 * i, xd + 4 * i);
            __builtin_amdgcn_s_wait_asynccnt(48);   // oldest 16 (x_t) complete
        } else {
            __builtin_amdgcn_s_wait_asynccnt(32);   // x_t complete; h in flight
        }
#else
        // Fallback: synchronous staging of x_t and h_{t-1}.
        {
            const float* xsrc = x + ((size_t)sb * S_LEN + t) * IN_DIM + sq * 64;
            float* ad = xA + (size_t)(t & 1) * XBUFSZ + sb * XSTRIDE + sq * 64;
            #pragma unroll
            for (int i = 0; i < 16; ++i)
                *(float4*)(ad + 4 * i) = *(const float4*)(xsrc + 4 * i);
            if (t + 1 < S_LEN) __builtin_prefetch(xsrc + IN_DIM, 0, 1);

            const float* hsrc = hbuf + (size_t)cur * (BATCH * HID) + sb * HID + sq * 128;
            float* ah = hA + sb * HSTRIDE + sq * 128;
            #pragma unroll
            for (int i = 0; i < 32; ++i)
                *(float4*)(ah + 4 * i) = *(const float4*)(hsrc + 4 * i);
        }
#endif
        __syncthreads();   // x_t visible to all waves

        v8f acc0 = {0.f,0.f,0.f,0.f,0.f,0.f,0.f,0.f};
        v8f acc1 = acc0, acc2 = acc0;
        const float* bp0 = bBase0;
        const float* bp1 = bBase1;
        const float* bp2 = bBase2;

        // ---- x-part K loop: 64 k-steps (independent of h) -----------------
        {
            const float* ap = xA + (size_t)(t & 1) * XBUFSZ
                              + (m0 + lrow) * XSTRIDE + koff;
            #pragma unroll 4
            for (int ks = 0; ks < KS_X; ++ks) {
                v2f av = *(const v2f*)ap;   ap  += 4;
                v2f b0 = *(const v2f*)bp0;  bp0 += 64;
                v2f b1 = *(const v2f*)bp1;  bp1 += 64;
                v2f b2 = *(const v2f*)bp2;  bp2 += 64;
                acc0 = __builtin_amdgcn_wmma_f32_16x16x4_f32(
                           false, av, false, b0, (short)0, acc0, false, false);
                acc1 = __builtin_amdgcn_wmma_f32_16x16x4_f32(
                           false, av, false, b1, (short)0, acc1, false, false);
                acc2 = __builtin_amdgcn_wmma_f32_16x16x4_f32(
                           false, av, false, b2, (short)0, acc2, false, false);
            }
        }

#if HAVE_ASYNC_LDS
        // h DMA must now be complete; only x_{t+1}(16) may remain in flight.
        if (t + 1 < S_LEN) __builtin_amdgcn_s_wait_asynccnt(16);
        else               __builtin_amdgcn_s_wait_asynccnt(0);
        __syncthreads();   // everyone's h rows visible
#endif

        // ---- h-part K loop: 128 k-steps ----------------------------------
        {
            const float* ap = aHBase;
            #pragma unroll 4
            for (int ks = 0; ks < KS_H; ++ks) {
                v2f av = *(const v2f*)ap;   ap  += 4;
                v2f b0 = *(const v2f*)bp0;  bp0 += 64;
                v2f b1 = *(const v2f*)bp1;  bp1 += 64;
                v2f b2 = *(const v2f*)bp2;  bp2 += 64;
                acc0 = __builtin_amdgcn_wmma_f32_16x16x4_f32(
                           false, av, false, b0, (short)0, acc0, false, false);
                acc1 = __builtin_amdgcn_wmma_f32_16x16x4_f32(
                           false, av, false, b1, (short)0, acc1, false, false);
                acc2 = __builtin_amdgcn_wmma_f32_16x16x4_f32(
                           false, av, false, b2, (short)0, acc2, false, false);
            }
        }

        // ---- Scatter gate tiles (+bias) into LDS for i/g/o coupling -------
        {
            const int rbase = m0 + ((lane < 16) ? 0 : 8);
            const int cl0 = (ntb + 0) * 16 + lrow;
            const int cl1 = (ntb + 1) * 16 + lrow;
            const int cl2 = (ntb + 2) * 16 + lrow;
            #pragma unroll
            for (int v = 0; v < 8; ++v) {
                gatesL[(rbase + v) * GSTRIDE + cl0] = acc0[v] + bias0;
                gatesL[(rbase + v) * GSTRIDE + cl1] = acc1[v] + bias1;
                gatesL[(rbase + v) * GSTRIDE + cl2] = acc2[v] + bias2;
            }
        }
        __syncthreads();

        // ---- Activations + h/c, write h to output + double buffer ---------
        {
            const float* gb = gatesL + sb * GSTRIDE;
            float* hdst = hbuf + (size_t)(cur ^ 1) * (BATCH * HID) + sb * HID;
            float* odst = out_hs + ((size_t)sb * S_LEN + t) * HID;
            #pragma unroll
            for (int j = 0; j < 8; ++j) {
                int c   = sq * 8 + j;                 // local h column 0..31
                int col = grp * HC + c;               // global h column
                float i_ = fsigmoid(gb[c]);
                float g_ = ftanh(gb[32 + c]);
                float o_ = fsigmoid(gb[64 + c]);
                float cc = i_ * g_;                   // no-forget cell state
                float hh = o_ * ftanh(cc);
                odst[col] = hh;
                hdst[col] = hh;
                if (t == S_LEN - 1) {
                    out_hf[sb * HID + col] = hh;
                    out_cf[sb * HID + col] = cc;
                }
            }
        }

        // ---- One grid-wide barrier per timestep ---------------------------
        if (t != S_LEN - 1) grid_sync(cnt, t);
        else                __syncthreads();
        cur ^= 1;
    }
}

// ---------------------------------------------------------------------------
// Host-side launch. Workspace layout:
//   [0,256)          grid barrier counter (zeroed each launch)
//   [256,262400)     h double buffer 2*64*512 f32 (zeroed each launch)
//   [262400,+4.5MB)  packed weight fragments Bpack (rewritten each launch)
// ---------------------------------------------------------------------------
extern "C" void kernel_launch(void* const* d_in, const int* in_sizes, int n_in,
                              void* d_out, int out_size, void* d_ws, size_t ws_size,
                              hipStream_t stream) {
    const float* x    = (const float*)d_in[0];
    const float* W    = (const float*)d_in[1];
    const float* U    = (const float*)d_in[2];
    const float* bias = (const float*)d_in[3];

    float* out_hs = (float*)d_out;
    float* out_hf = out_hs + (size_t)BATCH * S_LEN * HID;
    float* out_cf = out_hf + (size_t)BATCH * HID;

    unsigned* cnt = (unsigned*)d_ws;
    float* hbuf   = (float*)((char*)d_ws + 256);
    float* Bpack  = (float*)((char*)d_ws + 256 + (size_t)2 * BATCH * HID * sizeof(float));

    // Zero barrier counter + h0 (deterministic every call; capture-safe).
    (void)hipMemsetAsync(d_ws, 0, 256 + (size_t)2 * BATCH * HID * sizeof(float), stream);

    // Pack weights into WMMA fragment order (L2-resident, reread every step).
    const int packTot = NWG * NTILES * KSTEPS * 32;           // 589,824 threads
    pack_B_kernel<<<(packTot + 255) / 256, 256, 0, stream>>>(W, U, Bpack);

    // Persistent recurrence kernel: 16 WGs, 8 wave32 each, ~284 KB dynamic LDS.
    size_t dynLds = (size_t)(2 * XBUFSZ + BATCH * HSTRIDE + BATCH * GSTRIDE) * sizeof(float);
    lstm_noforget_kernel<<<NWG, 256, dynLds, stream>>>(
        x, bias, Bpack, hbuf, cnt, out_hs, out_hf, out_cf);
}